// Attention_59528246723073
// MI455X (gfx1250) — compile-verified
//
#include <hip/hip_runtime.h>
#include <stdint.h>

// Problem constants (from reference): B=32, C=128, S=256, T=64
#define NB 32
#define NC 128
#define NS 256
#define NT 64
#define SLICE (NS * NT)   // 16384 floats = 64 KB per (b,c) slice

typedef float v2f __attribute__((ext_vector_type(2)));
typedef float v8f __attribute__((ext_vector_type(8)));

// One block per (b,c) slice. 256 threads = 8 waves (wave32).
__global__ __launch_bounds__(256) void
attn_fused_kernel(const float* __restrict__ x,
                  const float* __restrict__ Wspect,
                  const float* __restrict__ Wtemp,
                  float* __restrict__ out) {
    __shared__ __align__(16) float xs[SLICE];   // x slice, row-major [S][T]
    __shared__ float s_wspect[NT];
    __shared__ float s_wtemp[NS];
    __shared__ float fs_raw[NS];
    __shared__ float ft_raw[NT];
    __shared__ float ss[NS];                    // softmax(fs) * 100
    __shared__ float st[NT];                    // softmax(ft)
    __shared__ float red[256];

    const int tid  = threadIdx.x;
    const int lane = tid & 31;
    const int wid  = tid >> 5;
    const int bc   = blockIdx.x;                // b*NC + c
    const int c    = bc & (NC - 1);

    const float* xslice = x + (size_t)bc * SLICE;

    // ---- Phase 1: async DMA x slice (64 KB) global -> LDS, 16B per lane per op
    // 4096 float4 chunks / 256 threads = 16 async b128 loads per thread.
    #pragma unroll
    for (int k = 0; k < 16; ++k) {
        int flat4 = k * 256 + tid;                         // float4 index
        uint32_t lds_off = (uint32_t)(uintptr_t)(&xs[flat4 * 4]);
        uint64_t gaddr   = (uint64_t)(uintptr_t)(xslice + flat4 * 4);
        asm volatile("global_load_async_to_lds_b128 %0, %1, off"
                     :: "v"(lds_off), "v"(gaddr) : "memory");
    }
    // weights via regular loads (tiny)
    if (tid < NT) s_wspect[tid] = Wspect[c * NT + tid];
    s_wtemp[tid] = Wtemp[c * NS + tid];
    asm volatile("s_wait_asynccnt 0" ::: "memory");
    __syncthreads();

    // ---- Phase 2a: fs[s] = sum_t xs[s][t] * wspect[t]  via V_WMMA_F32_16X16X4_F32
    // Each wave: rows [wid*32, wid*32+32) as two 16-row tiles; 16 K-chunks of 4.
    // A 16x4 f32 layout (ISA 7.12.2): v0 = K=0 (lanes 0-15) / K=2 (lanes 16-31),
    // v1 = K=1 / K=3. B's 4x16 columns are all identical (= wspect chunk), so
    // only B's K striping matters; it mirrors A's.
    {
        const int koff = (lane >= 16) ? 2 : 0;
        const int mrow = lane & 15;
        #pragma unroll
        for (int tile = 0; tile < 2; ++tile) {
            const int R = wid * 32 + tile * 16;
            const float* xrow = &xs[(R + mrow) * NT];
            v8f acc = {0.f, 0.f, 0.f, 0.f, 0.f, 0.f, 0.f, 0.f};
            #pragma unroll
            for (int k0 = 0; k0 < NT; k0 += 4) {
                v2f a, bm;
                a.x  = xrow[k0 + koff];
                a.y  = xrow[k0 + koff + 1];
                bm.x = s_wspect[k0 + koff];
                bm.y = s_wspect[k0 + koff + 1];
                acc = __builtin_amdgcn_wmma_f32_16x16x4_f32(
                    false, a, false, bm, (short)0, acc, false, false);
            }
            // Column N=0 lives in lane 0 (M=R..R+7) and lane 16 (M=R+8..R+15).
            if ((lane & 15) == 0) {
                const int base = R + ((lane >> 4) << 3);
                fs_raw[base + 0] = acc[0];
                fs_raw[base + 1] = acc[1];
                fs_raw[base + 2] = acc[2];
                fs_raw[base + 3] = acc[3];
                fs_raw[base + 4] = acc[4];
                fs_raw[base + 5] = acc[5];
                fs_raw[base + 6] = acc[6];
                fs_raw[base + 7] = acc[7];
            }
        }
    }

    // ---- Phase 2b: ft[t] = sum_s xs[s][t] * wtemp[s]  (VALU, 4 partials per t)
    {
        const int t = tid & (NT - 1);
        const int chunk = tid >> 6;            // 0..3, each covers 64 s values
        const float* wt = &s_wtemp[chunk * 64];
        const float* xc = &xs[(chunk * 64) * NT + t];
        float p = 0.f;
        #pragma unroll 8
        for (int j = 0; j < 64; ++j) p = fmaf(xc[j * NT], wt[j], p);
        red[tid] = p;
    }
    __syncthreads();
    if (tid < NT)
        ft_raw[tid] = red[tid] + red[tid + 64] + red[tid + 128] + red[tid + 192];
    __syncthreads();

    // ---- Phase 3a: softmax over S of tanh(fs)  (scaled by 100 here)
    {
        float v = tanhf(fs_raw[tid]);
        red[tid] = v; __syncthreads();
        for (int off = 128; off; off >>= 1) {
            if (tid < off) red[tid] = fmaxf(red[tid], red[tid + off]);
            __syncthreads();
        }
        float m = red[0]; __syncthreads();
        float e = expf(v - m);
        red[tid] = e; __syncthreads();
        for (int off = 128; off; off >>= 1) {
            if (tid < off) red[tid] += red[tid + off];
            __syncthreads();
        }
        float s = red[0]; __syncthreads();
        ss[tid] = 100.0f * e / s;
    }

    // ---- Phase 3b: softmax over T of tanh(ft) (mask inactive lanes)
    {
        float v = (tid < NT) ? tanhf(ft_raw[tid]) : -3.402823e38f;
        red[tid] = v; __syncthreads();
        for (int off = 128; off; off >>= 1) {
            if (tid < off) red[tid] = fmaxf(red[tid], red[tid + off]);
            __syncthreads();
        }
        float m = red[0]; __syncthreads();
        float e = (tid < NT) ? expf(v - m) : 0.f;
        red[tid] = e; __syncthreads();
        for (int off = 128; off; off >>= 1) {
            if (tid < off) red[tid] += red[tid + off];
            __syncthreads();
        }
        float s = red[0]; __syncthreads();
        if (tid < NT) st[tid] = e / s;
    }
    __syncthreads();

    // ---- Phase 4: out = x * ss[s] * st[t] (100 folded into ss), float4 stores
    float4* out4 = reinterpret_cast<float4*>(out + (size_t)bc * SLICE);
    #pragma unroll
    for (int k = 0; k < 16; ++k) {
        int flat4 = k * 256 + tid;
        int base  = flat4 * 4;
        int s     = base >> 6;          // /NT
        int t     = base & (NT - 1);
        float sm = ss[s];
        float4 v;
        v.x = xs[base + 0] * sm * st[t + 0];
        v.y = xs[base + 1] * sm * st[t + 1];
        v.z = xs[base + 2] * sm * st[t + 2];
        v.w = xs[base + 3] * sm * st[t + 3];
        out4[flat4] = v;
    }
}

extern "C" void kernel_launch(void* const* d_in, const int* in_sizes, int n_in,
                              void* d_out, int out_size, void* d_ws, size_t ws_size,
                              hipStream_t stream) {
    (void)in_sizes; (void)n_in; (void)d_ws; (void)ws_size; (void)out_size;
    const float* x      = (const float*)d_in[0];
    const float* Wspect = (const float*)d_in[1];
    const float* Wtemp  = (const float*)d_in[2];
    float* out          = (float*)d_out;
    dim3 grid(NB * NC);   // 4096 blocks, one per (b,c) slice
    dim3 block(256);      // 8 waves
    attn_fused_kernel<<<grid, block, 0, stream>>>(x, Wspect, Wtemp, out);
}